// SelectiveSSM_60181081752054
// MI455X (gfx1250) — compile-verified
//
#include <hip/hip_runtime.h>

typedef __attribute__((ext_vector_type(2))) float v2f;
typedef __attribute__((ext_vector_type(8))) float v8f;
typedef __attribute__((ext_vector_type(4))) int v4i;

#define BATCHN 32
#define SEQLEN 2048
#define DMODEL 512
#define CHLEN  32
#define NCHUNK 64   // SEQLEN / CHLEN

#ifndef __has_builtin
#define __has_builtin(x) 0
#endif
#if __has_builtin(__builtin_amdgcn_global_load_async_to_lds_b128) && \
    __has_builtin(__builtin_amdgcn_s_wait_asynccnt)
#define HAVE_ASYNC_LDS 1
#else
#define HAVE_ASYNC_LDS 0
#endif

#if HAVE_ASYNC_LDS
typedef __attribute__((address_space(1))) v4i* g4_t;   // global int4*
typedef __attribute__((address_space(3))) v4i* l4_t;   // LDS int4*
// Async DMA of 16B global -> LDS, tracked by ASYNCcnt (no VGPR round trip).
static __device__ __forceinline__ void async_ld16(const float* g, float* l) {
  __builtin_amdgcn_global_load_async_to_lds_b128(
      (g4_t)(uintptr_t)g, (l4_t)(unsigned)(uintptr_t)l, 0, 0);
}
#endif

static __device__ __forceinline__ v8f wmma4(v2f a, v2f b, v8f c) {
  // D(16x16,f32) = A(16x4,f32) * B(4x16,f32) + C
  return __builtin_amdgcn_wmma_f32_16x16x4_f32(false, a, false, b, (short)0, c,
                                               false, false);
}

// ---------------------------------------------------------------------------
// Precompute Mt[j][i] = A_bar[i][j] = exp(A_i * dt_j), transposed layout so the
// WMMA B-operand (K=i contiguous per output column j) is an 8-byte load.
__global__ void k_mt(const float* __restrict__ A_log,
                     const float* __restrict__ A_scl,
                     const float* __restrict__ dt_log,
                     float* __restrict__ Mt) {
  int idx = blockIdx.x * 256 + threadIdx.x;   // 512*512
  int j = idx >> 9;      // dt index (row of Mt)
  int i = idx & 511;     // state index
  float Ai  = -__expf(A_log[i] + A_scl[0]);
  float dtj = __expf(dt_log[j]);
  Mt[idx] = __expf(Ai * dtj);
}

// Bfac[n] = sum_e (exp(A_n*dt_e)-1)/A_n = (sum_j Mt[j][n] - 512) / A_n
__global__ void k_bfac(const float* __restrict__ Mt,
                       const float* __restrict__ A_log,
                       const float* __restrict__ A_scl,
                       float* __restrict__ Bfac) {
  int n = threadIdx.x;   // 512 threads
  float s = 0.f;
  for (int j = 0; j < DMODEL; ++j) s += Mt[j * 512 + n];
  float An = -__expf(A_log[n] + A_scl[0]);
  Bfac[n] = (s - (float)DMODEL) / An;
}

// W1t[n][d] = B[d][n] * Bfac[n]   (transposed, Bfac folded in)
__global__ void k_w1t(const float* __restrict__ Bm,
                      const float* __restrict__ Bfac,
                      float* __restrict__ W1t) {
  int idx = blockIdx.x * 256 + threadIdx.x;   // 512*512
  int n = idx >> 9;
  int d = idx & 511;
  W1t[idx] = Bm[d * 512 + n] * Bfac[n];
}

// ---------------------------------------------------------------------------
// Fused dual GEMM: X = u @ W1 (u_B_sum) and UC = u @ C^T, both 65536x512, K=512.
// Block = 256 threads (8 waves), tile 128 rows x 64 cols, K staged in 32-chunks.
// Grid is (cols, rows) so the 8 column-blocks sharing a u row-panel are
// dispatch-adjacent and the panel stays resident in the 192MB L2.
__global__ __launch_bounds__(256)
void k_gemm_xuc(const float* __restrict__ u, const float* __restrict__ W1t,
                const float* __restrict__ Cm, float* __restrict__ X,
                float* __restrict__ UC) {
  __shared__ __attribute__((aligned(16))) float sU[128 * 36];
  __shared__ __attribute__((aligned(16))) float sW[64 * 36];
  __shared__ __attribute__((aligned(16))) float sC[64 * 36];
  const int tid  = threadIdx.x;
  const int wave = tid >> 5;
  const int lane = tid & 31;
  const int lm   = lane & 15;
  const int hs   = lane >> 4;
  const int colBase = blockIdx.x * 64;
  const int rowBase = blockIdx.y * 128;

  v8f z = {0.f, 0.f, 0.f, 0.f, 0.f, 0.f, 0.f, 0.f};
  v8f accX[4], accU[4];
#pragma unroll
  for (int i = 0; i < 4; ++i) { accX[i] = z; accU[i] = z; }

  for (int k0 = 0; k0 < 512; k0 += 32) {
    __syncthreads();   // all LDS readers from previous chunk done
#if HAVE_ASYNC_LDS
#pragma unroll
    for (int i = 0; i < 4; ++i) {              // U tile: 128x32 = 1024 x 16B
      int idx = tid + i * 256;
      int r = idx >> 3, c4 = (idx & 7) * 4;
      async_ld16(&u[(size_t)(rowBase + r) * 512 + k0 + c4], &sU[r * 36 + c4]);
    }
#pragma unroll
    for (int i = 0; i < 2; ++i) {              // W / C tiles: 64x32 each
      int idx = tid + i * 256;
      int r = idx >> 3, c4 = (idx & 7) * 4;
      async_ld16(&W1t[(size_t)(colBase + r) * 512 + k0 + c4], &sW[r * 36 + c4]);
      async_ld16(&Cm[(size_t)(colBase + r) * 512 + k0 + c4], &sC[r * 36 + c4]);
    }
    __builtin_amdgcn_s_wait_asynccnt(0);
#else
#pragma unroll
    for (int i = 0; i < 4; ++i) {
      int idx = tid + i * 256;
      int r = idx >> 3, c4 = (idx & 7) * 4;
      *(float4*)&sU[r * 36 + c4] =
          *(const float4*)&u[(size_t)(rowBase + r) * 512 + k0 + c4];
    }
#pragma unroll
    for (int i = 0; i < 2; ++i) {
      int idx = tid + i * 256;
      int r = idx >> 3, c4 = (idx & 7) * 4;
      *(float4*)&sW[r * 36 + c4] =
          *(const float4*)&W1t[(size_t)(colBase + r) * 512 + k0 + c4];
      *(float4*)&sC[r * 36 + c4] =
          *(const float4*)&Cm[(size_t)(colBase + r) * 512 + k0 + c4];
    }
#endif
    __syncthreads();
    for (int kk = 0; kk < 32; kk += 4) {
      v2f a = *(const v2f*)&sU[(wave * 16 + lm) * 36 + kk + 2 * hs];
#pragma unroll
      for (int ct = 0; ct < 4; ++ct) {
        v2f bw = *(const v2f*)&sW[(ct * 16 + lm) * 36 + kk + 2 * hs];
        accX[ct] = wmma4(a, bw, accX[ct]);
        v2f bc = *(const v2f*)&sC[(ct * 16 + lm) * 36 + kk + 2 * hs];
        accU[ct] = wmma4(a, bc, accU[ct]);
      }
    }
  }
  const int m0 = rowBase + wave * 16;
#pragma unroll
  for (int ct = 0; ct < 4; ++ct) {
    int n0 = colBase + ct * 16;
#pragma unroll
    for (int r = 0; r < 8; ++r) {
      size_t row = (size_t)(m0 + r + 8 * hs);
      X[row * 512 + n0 + lm]  = accX[ct][r];
      UC[row * 512 + n0 + lm] = accU[ct][r];
    }
  }
}

// ---------------------------------------------------------------------------
// Matrix squaring in transposed layout: Out = In @ In (row-major on Mt gives
// (M^2)^T). Tiny (134 MFLOP, L2-resident, coalesced) — per-element kernel.
__global__ void k_square(const float* __restrict__ In, float* __restrict__ Out) {
  int idx = blockIdx.x * 256 + threadIdx.x;   // 512*512
  int r = idx >> 9, c = idx & 511;
  float s = 0.f;
  for (int k = 0; k < 512; ++k) s += In[r * 512 + k] * In[k * 512 + c];
  Out[idx] = s;
}

// ---------------------------------------------------------------------------
// Chunked scan. One workgroup (512 thr = 16 waves) per chunk of 32 timesteps.
// Each step: H(32x512) = H @ M + X_t via WMMA, H resident in LDS (stride 514
// spreads the 16 A-operand rows across distinct bank pairs).
// phase1 (phase3==0): H starts at 0, writes final local state S_chunk.
// phase3 (phase3==1): H starts at Hin[chunk], also emits y[b,t] = <h_t, uC_t>.
__global__ __launch_bounds__(512)
void k_scan(const float* __restrict__ X, const float* __restrict__ Mt,
            const float* __restrict__ UC, const float* __restrict__ Hin,
            float* __restrict__ Sout, float* __restrict__ y, int phase3) {
  extern __shared__ float smem[];
  float* H     = smem;               // 32 x 514
  float* yPart = smem + 32 * 514;    // 32 x 16 deterministic partials
  const int tid  = threadIdx.x;
  const int wave = tid >> 5;
  const int lane = tid & 31;
  const int lm   = lane & 15;
  const int hs   = lane >> 4;
  const int chunk = blockIdx.x;
  const int t0 = chunk * CHLEN;
  const int nw0 = wave * 32;         // this wave owns columns nw0..nw0+31

  for (int i = tid; i < 32 * 512; i += 512) {
    int b = i >> 9, n = i & 511;
    H[b * 514 + n] = phase3 ? Hin[(size_t)chunk * (32 * 512) + i] : 0.f;
  }
  __syncthreads();

  for (int k = 0; k < CHLEN; ++k) {
    const int t = t0 + k;
    v8f acc[2][2];
#pragma unroll
    for (int mt = 0; mt < 2; ++mt)
#pragma unroll
      for (int ct = 0; ct < 2; ++ct) {
        int n0 = nw0 + ct * 16;
#pragma unroll
        for (int r = 0; r < 8; ++r) {
          int b = mt * 16 + r + 8 * hs;
          acc[mt][ct][r] = X[((size_t)b * SEQLEN + t) * 512 + n0 + lm];
        }
      }
    if (k + 1 < CHLEN)   // warm next step's x-tile into cache
      __builtin_prefetch(&X[((size_t)(8 * hs) * SEQLEN + t + 1) * 512 + nw0 + lm], 0, 1);

    for (int kk = 0; kk < 512; kk += 4) {
      v2f a0 = *(const v2f*)&H[lm * 514 + kk + 2 * hs];
      v2f a1 = *(const v2f*)&H[(16 + lm) * 514 + kk + 2 * hs];
#pragma unroll
      for (int ct = 0; ct < 2; ++ct) {
        v2f b = *(const v2f*)&Mt[(size_t)(nw0 + ct * 16 + lm) * 512 + kk + 2 * hs];
        acc[0][ct] = wmma4(a0, b, acc[0][ct]);
        acc[1][ct] = wmma4(a1, b, acc[1][ct]);
      }
    }
    __syncthreads();                 // all reads of H complete
#pragma unroll
    for (int mt = 0; mt < 2; ++mt)
#pragma unroll
      for (int ct = 0; ct < 2; ++ct) {
        int n0 = nw0 + ct * 16;
#pragma unroll
        for (int r = 0; r < 8; ++r) {
          int b = mt * 16 + r + 8 * hs;
          H[b * 514 + n0 + lm] = acc[mt][ct][r];
        }
      }
    __syncthreads();

    if (phase3) {
      int b = tid >> 4, s = tid & 15, nb = s * 32;
      const float* hrow  = &H[b * 514 + nb];                       // 8B aligned
      const float* ucrow = &UC[((size_t)b * SEQLEN + t) * 512 + nb]; // 16B aligned
      float p = 0.f;
#pragma unroll
      for (int i = 0; i < 32; i += 4) {
        float4 uc4 = *(const float4*)&ucrow[i];
        float2 h0  = *(const float2*)&hrow[i];
        float2 h1  = *(const float2*)&hrow[i + 2];
        p += h0.x * uc4.x + h0.y * uc4.y + h1.x * uc4.z + h1.y * uc4.w;
      }
      yPart[b * 16 + s] = p;
      __syncthreads();
      if (tid < 32) {
        float sum = 0.f;
#pragma unroll
        for (int i = 0; i < 16; ++i) sum += yPart[tid * 16 + i];
        y[(size_t)tid * SEQLEN + t] = sum;
      }
      __syncthreads();
    }
  }

  if (!phase3) {
    for (int i = tid; i < 32 * 512; i += 512) {
      int b = i >> 9, n = i & 511;
      Sout[(size_t)chunk * (32 * 512) + i] = H[b * 514 + n];
    }
  }
}

// ---------------------------------------------------------------------------
// Sequential combine across 64 chunks (single WG): Hin_{j+1} = Hin_j @ P + S_j,
// with P = M^32 (Pt in transposed layout). Writes every chunk-entry state.
__global__ __launch_bounds__(512)
void k_combine(const float* __restrict__ S, const float* __restrict__ Pt,
               float* __restrict__ Hin) {
  extern __shared__ float smem[];
  float* H = smem;                   // 32 x 514
  const int tid  = threadIdx.x;
  const int wave = tid >> 5;
  const int lane = tid & 31;
  const int lm   = lane & 15;
  const int hs   = lane >> 4;
  const int nw0  = wave * 32;

  for (int i = tid; i < 32 * 512; i += 512) {
    int b = i >> 9, n = i & 511;
    H[b * 514 + n] = 0.f;
  }
  __syncthreads();

  for (int j = 0; j < NCHUNK; ++j) {
    for (int i = tid; i < 32 * 512; i += 512) {   // publish state entering chunk j
      int b = i >> 9, n = i & 511;
      Hin[(size_t)j * (32 * 512) + i] = H[b * 514 + n];
    }
    v8f acc[2][2];
#pragma unroll
    for (int mt = 0; mt < 2; ++mt)
#pragma unroll
      for (int ct = 0; ct < 2; ++ct) {
        int n0 = nw0 + ct * 16;
#pragma unroll
        for (int r = 0; r < 8; ++r) {
          int b = mt * 16 + r + 8 * hs;
          acc[mt][ct][r] = S[(size_t)j * (32 * 512) + b * 512 + n0 + lm];
        }
      }
    for (int kk = 0; kk < 512; kk += 4) {
      v2f a0 = *(const v2f*)&H[lm * 514 + kk + 2 * hs];
      v2f a1 = *(const v2f*)&H[(16 + lm) * 514 + kk + 2 * hs];
#pragma unroll
      for (int ct = 0; ct < 2; ++ct) {
        v2f b = *(const v2f*)&Pt[(size_t)(nw0 + ct * 16 + lm) * 512 + kk + 2 * hs];
        acc[0][ct] = wmma4(a0, b, acc[0][ct]);
        acc[1][ct] = wmma4(a1, b, acc[1][ct]);
      }
    }
    __syncthreads();
#pragma unroll
    for (int mt = 0; mt < 2; ++mt)
#pragma unroll
      for (int ct = 0; ct < 2; ++ct) {
        int n0 = nw0 + ct * 16;
#pragma unroll
        for (int r = 0; r < 8; ++r) {
          int b = mt * 16 + r + 8 * hs;
          H[b * 514 + n0 + lm] = acc[mt][ct][r];
        }
      }
    __syncthreads();
  }
}

// out[b,l,d] = y[b,l] + u[b,l,d] * D[d]
__global__ void k_final(const float* __restrict__ u, const float* __restrict__ Dv,
                        const float* __restrict__ y, float* __restrict__ out) {
  size_t idx = (size_t)blockIdx.x * 256 + threadIdx.x;
  int d = (int)(idx & 511);
  size_t bt = idx >> 9;
  out[idx] = y[bt] + u[idx] * Dv[d];
}

// ---------------------------------------------------------------------------
extern "C" void kernel_launch(void* const* d_in, const int* in_sizes, int n_in,
                              void* d_out, int out_size, void* d_ws, size_t ws_size,
                              hipStream_t stream) {
  const float* u      = (const float*)d_in[0];
  const float* A_log  = (const float*)d_in[1];
  const float* A_scl  = (const float*)d_in[2];
  const float* Bm     = (const float*)d_in[3];
  const float* Cm     = (const float*)d_in[4];
  const float* dt_log = (const float*)d_in[5];
  const float* Dv     = (const float*)d_in[6];
  float* out = (float*)d_out;
  float* ws  = (float*)d_ws;

  const size_t off_Mt   = 0;
  const size_t off_P0   = off_Mt  + (size_t)512 * 512;
  const size_t off_P1   = off_P0  + (size_t)512 * 512;
  const size_t off_W1t  = off_P1  + (size_t)512 * 512;
  const size_t off_Bfac = off_W1t + (size_t)512 * 512;
  const size_t off_y    = off_Bfac + 512;
  const size_t off_S    = off_y   + (size_t)BATCHN * SEQLEN;
  const size_t off_Hin  = off_S   + (size_t)NCHUNK * BATCHN * DMODEL;
  const size_t off_X    = off_Hin + (size_t)NCHUNK * BATCHN * DMODEL;
  const size_t off_UC   = off_X   + (size_t)BATCHN * SEQLEN * DMODEL;

  float* Mt  = ws + off_Mt;
  float* P0  = ws + off_P0;
  float* P1  = ws + off_P1;
  float* W1t = ws + off_W1t;
  float* Bf  = ws + off_Bfac;
  float* yv  = ws + off_y;
  float* S   = ws + off_S;
  float* Hin = ws + off_Hin;
  float* X   = ws + off_X;
  float* UC  = ws + off_UC;

  const size_t smemScan = (32 * 514 + 32 * 16) * sizeof(float);
  const size_t smemComb = (32 * 514) * sizeof(float);

  // Precompute discretization (transposed layouts for WMMA B-operands)
  k_mt  <<<1024, 256, 0, stream>>>(A_log, A_scl, dt_log, Mt);
  k_bfac<<<1,    512, 0, stream>>>(Mt, A_log, A_scl, Bf);
  k_w1t <<<1024, 256, 0, stream>>>(Bm, Bf, W1t);

  // Dual GEMM: X = u@W1, UC = u@C^T  (grid = (cols, rows) for L2 reuse of u)
  k_gemm_xuc<<<dim3(8, 512), 256, 0, stream>>>(u, W1t, Cm, X, UC);

  // P = M^32 via 5 squarings (ends in P0)
  k_square<<<1024, 256, 0, stream>>>(Mt, P0);
  k_square<<<1024, 256, 0, stream>>>(P0, P1);
  k_square<<<1024, 256, 0, stream>>>(P1, P0);
  k_square<<<1024, 256, 0, stream>>>(P0, P1);
  k_square<<<1024, 256, 0, stream>>>(P1, P0);

  // Phase 1: local chunk scans (64-way parallel) -> S
  k_scan<<<NCHUNK, 512, smemScan, stream>>>(X, Mt, nullptr, nullptr, S, nullptr, 0);
  // Phase 2: sequential combine -> Hin
  k_combine<<<1, 512, smemComb, stream>>>(S, P0, Hin);
  // Phase 3: re-scan with correct entry states, emit y
  k_scan<<<NCHUNK, 512, smemScan, stream>>>(X, Mt, UC, Hin, nullptr, yv, 1);

  // Epilogue: broadcast y + skip connection
  k_final<<<131072, 256, 0, stream>>>(u, Dv, yv, out);
}